// PostProcessor_16269336118081
// MI455X (gfx1250) — compile-verified
//
#include <hip/hip_runtime.h>
#include <math.h>

// Problem constants (from reference)
#define BB 32
#define NN 128
#define CC 151
#define PP 51
#define RR 4096
#define THRESH 0.3f

// d_out layout (flat concat, reference return order, all as float):
//  boxes_out : BB*NN*4      @ 0
//  obj_pred  : BB*NN        @ 16384
//  obj_scores: BB*NN        @ 20480
//  pair_sort : BB*RR*2      @ 24576
//  prob_sort : BB*RR*PP     @ 286720
//  class_sort: BB*RR        @ 6971392
#define O_BOXES  0
#define O_PRED   16384
#define O_SCORES 20480
#define O_PAIR   24576
#define O_PROB   286720
#define O_CLASS  6971392

typedef __attribute__((ext_vector_type(2))) float v2f;
typedef __attribute__((ext_vector_type(8))) float v8f;

// ---------------------------------------------------------------------------
// Kernel 1: per-batch object softmax + greedy class-aware NMS (128 serial steps)
// One block (256 threads) per batch element. prob matrix lives in LDS.
// ---------------------------------------------------------------------------
__global__ void __launch_bounds__(256)
objnms_kernel(const float* __restrict__ obj_logits,   // (B,N,C)
              const float* __restrict__ boxes,        // (B,N,C,4)
              float* __restrict__ out,
              float* __restrict__ wsObjScores)        // (B,N)
{
    __shared__ float prob[NN * CC];        // 77312 B
    __shared__ float rmax[NN], rsum[NN];
    __shared__ int   labels[NN];
    __shared__ float redv[256];
    __shared__ int   redi[256];
    __shared__ int   sel_bi, sel_ci;

    const int b   = blockIdx.x;
    const int tid = threadIdx.x;
    const float* L = obj_logits + (size_t)b * NN * CC;

    // stable softmax per row, zero class 0; remember (max,sum) to recover
    // the *unmutated* obj_prob later for obj_scores.
    if (tid < NN) {
        const float* row = L + tid * CC;
        float m = -3.4e38f;
        for (int c = 0; c < CC; ++c) m = fmaxf(m, row[c]);
        float s = 0.f;
        for (int c = 0; c < CC; ++c) {
            float e = __expf(row[c] - m);
            prob[tid * CC + c] = e;
            s += e;
        }
        float inv = 1.f / s;
        for (int c = 0; c < CC; ++c) prob[tid * CC + c] *= inv;
        prob[tid * CC + 0] = 0.f;
        rmax[tid] = m; rsum[tid] = s;
        labels[tid] = 0;
    }
    __syncthreads();

    // 128 sequential greedy-NMS steps
    for (int t = 0; t < NN; ++t) {
        // flat argmax with first-index tie-break (== jnp.argmax)
        float bv = -3.4e38f; int bidx = 0;
        for (int i = tid; i < NN * CC; i += 256) {
            float v = prob[i];
            if (v > bv) { bv = v; bidx = i; }
        }
        redv[tid] = bv; redi[tid] = bidx;
        __syncthreads();
        for (int s = 128; s >= 1; s >>= 1) {
            if (tid < s) {
                float vo = redv[tid + s]; int io = redi[tid + s];
                if (vo > redv[tid] || (vo == redv[tid] && io < redi[tid])) {
                    redv[tid] = vo; redi[tid] = io;
                }
            }
            __syncthreads();
        }
        if (tid == 0) {
            int f = redi[0];
            sel_bi = f / CC; sel_ci = f - (f / CC) * CC;
            labels[sel_bi] = sel_ci;
        }
        __syncthreads();
        const int bi = sel_bi, ci = sel_ci;

        // suppress column ci where IoU(sel, j) >= THRESH (self included)
        if (tid < NN) {
            const float* bs = boxes + (((size_t)b * NN + bi) * CC + ci) * 4;
            const float* bj = boxes + (((size_t)b * NN + tid) * CC + ci) * 4;
            float ax1 = bs[0], ay1 = bs[1], ax2 = bs[2], ay2 = bs[3];
            float bx1 = bj[0], by1 = bj[1], bx2 = bj[2], by2 = bj[3];
            float iw = fmaxf(fminf(ax2, bx2) - fmaxf(ax1, bx1) + 1.f, 0.f);
            float ih = fmaxf(fminf(ay2, by2) - fmaxf(ay1, by1) + 1.f, 0.f);
            float inter = iw * ih;
            float aa = (ax2 - ax1 + 1.f) * (ay2 - ay1 + 1.f);
            float ab = (bx2 - bx1 + 1.f) * (by2 - by1 + 1.f);
            float iou = inter / (aa + ab - inter);
            if (iou >= THRESH) prob[tid * CC + ci] = 0.f;
        }
        __syncthreads();
        if (tid < CC) prob[bi * CC + tid] = -1.f;   // row bi <- -1
        __syncthreads();
    }

    // outputs: boxes_out, obj_pred, obj_scores (recovered from logits)
    if (tid < NN) {
        int lab = labels[tid];
        float sc = 0.f;
        if (lab > 0)
            sc = __expf(L[tid * CC + lab] - rmax[tid]) / rsum[tid];
        out[O_PRED   + b * NN + tid] = (float)lab;
        out[O_SCORES + b * NN + tid] = sc;
        wsObjScores[b * NN + tid]    = sc;
        const float* bb = boxes + (((size_t)b * NN + tid) * CC + lab) * 4;
        float* ob = out + O_BOXES + ((size_t)b * NN + tid) * 4;
        ob[0] = bb[0]; ob[1] = bb[1]; ob[2] = bb[2]; ob[3] = bb[3];
    }
}

// ---------------------------------------------------------------------------
// Kernel 2: rel softmax denominators via V_WMMA_F32_16X16X4_F32 row-sums,
// per-row max/argmax over classes 1..50, triple keys. One block per batch,
// 8 waves x 32 tiles of 16 rows each.
// ---------------------------------------------------------------------------
__global__ void __launch_bounds__(256)
relkeys_kernel(const float* __restrict__ rel_logits,  // (B,R,P)
               const int*   __restrict__ pair_idx,    // (B,R,2)
               const float* __restrict__ objScores,   // (B,N)
               float* __restrict__ wsKeys,            // (B,R)
               int*   __restrict__ wsClass)           // (B,R)
{
    const int b    = blockIdx.x;
    const int wave = threadIdx.x >> 5;
    const int lane = threadIdx.x & 31;
    const int half = lane >> 4;        // 0: K=0,1   1: K=2,3 (A 16x4 layout)
    const int lr   = lane & 15;        // matrix row within tile

    v2f ones; ones[0] = 1.f; ones[1] = 1.f;

    for (int tile = wave; tile < RR / 16; tile += 8) {
        const int r0 = tile * 16;
        const float* Lt   = rel_logits + ((size_t)b * RR + r0) * PP;
        const float* rowp = Lt + lr * PP;
        __builtin_prefetch(rowp, 0, 1);           // global_prefetch_b8

        // D = sum_k exp(A) * ones : exact f32 row sums over K=51 (13 chunks)
        v8f acc = {};
        #pragma unroll
        for (int kc = 0; kc < 13; ++kc) {
            int k0 = kc * 4 + half * 2;
            v2f a;
            a[0] = (k0     < PP) ? __expf(rowp[k0])     : 0.f;
            a[1] = (k0 + 1 < PP) ? __expf(rowp[k0 + 1]) : 0.f;
            acc = __builtin_amdgcn_wmma_f32_16x16x4_f32(
                false, a, false, ones, (short)0, acc, false, false);
        }

        // extract row sums: rows 0..7 live in lane0 v[0..7], rows 8..15 in lane16
        float rowsum = 0.f;
        #pragma unroll
        for (int m = 0; m < 8; ++m) {
            float sA = __shfl(acc[m], 0, 32);
            float sB = __shfl(acc[m], 16, 32);
            if (lr == m)     rowsum = sA;
            if (lr == m + 8) rowsum = sB;
        }

        if (lane < 16) {                 // one lane per row of the tile
            float best = -1.f; int bc = 1;
            for (int c = 1; c < PP; ++c) {
                float e = __expf(rowp[c]);
                if (e > best) { best = e; bc = c; }   // first-max tie-break
            }
            int row = r0 + lr;
            float score = best / rowsum;
            int p0 = pair_idx[((size_t)b * RR + row) * 2 + 0];
            int p1 = pair_idx[((size_t)b * RR + row) * 2 + 1];
            float key = score * objScores[b * NN + p0] * objScores[b * NN + p1];
            wsKeys[(size_t)b * RR + row]  = key;
            wsClass[(size_t)b * RR + row] = bc;
        }
    }
}

// ---------------------------------------------------------------------------
// Kernel 3: per-batch bitonic sort of 4096 (key,idx), descending key with
// ascending-index tie-break (== stable argsort of -key). Writes sorted
// pair_idx + rel_class, and order[] for kernel 4.
// ---------------------------------------------------------------------------
__global__ void __launch_bounds__(256)
relsort_kernel(const float* __restrict__ wsKeys,
               const int*   __restrict__ wsClass,
               const int*   __restrict__ pair_idx,
               float* __restrict__ out,
               int*   __restrict__ wsOrder)
{
    __shared__ float sk[RR];
    __shared__ int   si[RR];
    const int b   = blockIdx.x;
    const int tid = threadIdx.x;

    for (int i = tid; i < RR; i += 256) {
        sk[i] = wsKeys[(size_t)b * RR + i];
        si[i] = i;
    }
    __syncthreads();

    for (int k = 2; k <= RR; k <<= 1) {
        for (int j = k >> 1; j > 0; j >>= 1) {
            for (int i = tid; i < RR; i += 256) {
                int p = i ^ j;
                if (p > i) {
                    float ka = sk[i], kb = sk[p];
                    int   ia = si[i], ib = si[p];
                    bool aBeforeB = (ka > kb) || (ka == kb && ia < ib);
                    bool up = ((i & k) == 0);        // descending region
                    if (up ? !aBeforeB : aBeforeB) {
                        sk[i] = kb; sk[p] = ka;
                        si[i] = ib; si[p] = ia;
                    }
                }
            }
            __syncthreads();
        }
    }

    for (int i = tid; i < RR; i += 256) {
        int src = si[i];
        wsOrder[(size_t)b * RR + i] = src;
        out[O_PAIR + ((size_t)b * RR + i) * 2 + 0] =
            (float)pair_idx[((size_t)b * RR + src) * 2 + 0];
        out[O_PAIR + ((size_t)b * RR + i) * 2 + 1] =
            (float)pair_idx[((size_t)b * RR + src) * 2 + 1];
        out[O_CLASS + (size_t)b * RR + i] = (float)wsClass[(size_t)b * RR + src];
    }
}

// ---------------------------------------------------------------------------
// Kernel 4: gather rel_prob[order] by recomputing the source row's softmax
// (avoids a 27MB scratch). One wave32 per output row.
// ---------------------------------------------------------------------------
__global__ void __launch_bounds__(256)
relgather_kernel(const float* __restrict__ rel_logits,
                 const int*   __restrict__ wsOrder,
                 float* __restrict__ out)
{
    const int b    = blockIdx.y;
    const int orow = blockIdx.x * 8 + (threadIdx.x >> 5);
    const int lane = threadIdx.x & 31;
    const int src  = wsOrder[(size_t)b * RR + orow];
    const float* rp = rel_logits + ((size_t)b * RR + src) * PP;

    float e0 = __expf(rp[lane]);                            // lanes 0..31 < 51
    float e1 = (lane + 32 < PP) ? __expf(rp[lane + 32]) : 0.f;
    float s = e0 + e1;
    #pragma unroll
    for (int off = 16; off > 0; off >>= 1) s += __shfl_xor(s, off, 32);
    float inv = 1.f / s;

    float* op = out + O_PROB + ((size_t)b * RR + orow) * PP;
    op[lane] = e0 * inv;
    if (lane + 32 < PP) op[lane + 32] = e1 * inv;
}

// ---------------------------------------------------------------------------
extern "C" void kernel_launch(void* const* d_in, const int* in_sizes, int n_in,
                              void* d_out, int out_size, void* d_ws, size_t ws_size,
                              hipStream_t stream) {
    const float* rel   = (const float*)d_in[0];  // (B,R,P)
    const float* obj   = (const float*)d_in[1];  // (B,N,C)
    const float* boxes = (const float*)d_in[2];  // (B,N,C,4)
    const int*   pair  = (const int*)d_in[3];    // (B,R,2)
    float* out = (float*)d_out;

    // ws: objScores[B*N] f32 | keys[B*R] f32 | class[B*R] i32 | order[B*R] i32
    float* wsObj   = (float*)d_ws;
    float* wsKeys  = wsObj + BB * NN;
    int*   wsClass = (int*)(wsKeys + (size_t)BB * RR);
    int*   wsOrder = wsClass + (size_t)BB * RR;

    objnms_kernel<<<BB, 256, 0, stream>>>(obj, boxes, out, wsObj);
    relkeys_kernel<<<BB, 256, 0, stream>>>(rel, pair, wsObj, wsKeys, wsClass);
    relsort_kernel<<<BB, 256, 0, stream>>>(wsKeys, wsClass, pair, out, wsOrder);
    relgather_kernel<<<dim3(RR / 8, BB), 256, 0, stream>>>(rel, wsOrder, out);
}